// Net_90074054132249
// MI455X (gfx1250) — compile-verified
//
#include <hip/hip_runtime.h>
#include <hip/hip_bf16.h>

typedef __attribute__((ext_vector_type(2))) float v2f;
typedef __attribute__((ext_vector_type(8))) float v8f;

// ---------------------------------------------------------------------------
// Zero-fill (n4 = number of float4 chunks)
// ---------------------------------------------------------------------------
__global__ void k_fill_zero(float* __restrict__ p, long n4) {
    long i = (long)blockIdx.x * blockDim.x + threadIdx.x;
    if (i < n4) {
        *(float4*)(p + i * 4) = make_float4(0.f, 0.f, 0.f, 0.f);
    }
}

// ---------------------------------------------------------------------------
// In-degree count (float, exact up to 2^24) and reciprocal
// ---------------------------------------------------------------------------
__global__ void k_count_deg(const int* __restrict__ ei, int E, float* __restrict__ cnt) {
    int e = blockIdx.x * blockDim.x + threadIdx.x;
    if (e < E) {
        int d = ei[E + e];                 // dst row
        unsafeAtomicAdd(&cnt[d], 1.0f);    // -> global_atomic_add_f32
    }
}

__global__ void k_inv_deg(const float* __restrict__ cnt, float* __restrict__ inv, int Nn) {
    int i = blockIdx.x * blockDim.x + threadIdx.x;
    if (i < Nn) inv[i] = 1.0f / fmaxf(cnt[i], 1.0f);
}

// ---------------------------------------------------------------------------
// Zero-pad a din x dout weight into din x doutp (doutp multiple of 64, pow2)
// ---------------------------------------------------------------------------
__global__ void k_pack_pad(const float* __restrict__ W, float* __restrict__ Wp,
                           int din, int dout, int doutp) {
    int i = blockIdx.x * blockDim.x + threadIdx.x;
    int total = din * doutp;
    if (i >= total) return;
    int k = i / doutp;               // doutp is a power of two -> cheap
    int n = i - k * doutp;
    Wp[i] = (n < dout) ? W[k * dout + n] : 0.f;
}

__global__ void k_pack_bias(const float* __restrict__ b, float* __restrict__ bp,
                            int dout, int doutp) {
    int i = blockIdx.x * blockDim.x + threadIdx.x;
    if (i < doutp) bp[i] = (i < dout) ? b[i] : 0.f;
}

// ---------------------------------------------------------------------------
// Edge scatter: agg[dst,:] += h[src,:]  (din multiple of 4)
// Consecutive threads cover consecutive 4-float chunks of one edge -> coalesced
// ---------------------------------------------------------------------------
__global__ void k_scatter(const int* __restrict__ ei, int E,
                          const float* __restrict__ h, float* __restrict__ agg,
                          int din, int total /* = E * (din/4) */) {
    int t = blockIdx.x * blockDim.x + threadIdx.x;
    if (t >= total) return;
    int chunks = din >> 2;
    int e = t / chunks;
    int c = (t - e * chunks) << 2;
    int s = ei[e];
    int d = ei[E + e];
    const float4 v = *(const float4*)(h + (long)s * din + c);
    float* p = agg + (long)d * din + c;
    unsafeAtomicAdd(p + 0, v.x);
    unsafeAtomicAdd(p + 1, v.y);
    unsafeAtomicAdd(p + 2, v.z);
    unsafeAtomicAdd(p + 3, v.w);
}

// ---------------------------------------------------------------------------
// Fused SAGE layer GEMM:
//   out = [relu]( (agg * inv_cnt) @ Wl + b + h @ Wr )
// One wave computes a 16(M) x 64(N) tile using V_WMMA_F32_16X16X4_F32.
// Weights/bias are pre-padded to doutp (multiple of 64) so the K loop has
// NO conditionals: 2x b64 A-loads, 16x coalesced b32 B-loads, 8 WMMAs.
// Column masking only in the store epilogue (dout=500 case).
// ---------------------------------------------------------------------------
__global__ void k_sage_gemm(const float* __restrict__ h,
                            const float* __restrict__ agg,
                            const float* __restrict__ inv,
                            const float* __restrict__ Wl,   // din x doutp, padded
                            const float* __restrict__ Wr,   // din x doutp, padded
                            const float* __restrict__ bias, // doutp, padded
                            float* __restrict__ out,        // Nn x dout (unpadded)
                            int Nn, int din, int dout, int doutp,
                            int mtiles, int ntiles, int do_relu) {
    const int lane = threadIdx.x & 31;
    const int wv   = threadIdx.x >> 5;
    const int gw   = blockIdx.x * (blockDim.x >> 5) + wv;
    if (gw >= mtiles * ntiles) return;

    const int mt = gw % mtiles;
    const int nt = gw / mtiles;
    const int m0 = mt << 4;
    const int n0 = nt << 6;

    const int r  = lane & 15;   // row within A tile / col within B,C tiles
    const int hi = lane >> 4;   // K-half select

    int arow_i = m0 + r;
    if (arow_i >= Nn) arow_i = Nn - 1;           // safe clamp (Nn%16==0 normally)
    const float* hrow = h   + (long)arow_i * din + 2 * hi;
    const float* arow = agg + (long)arow_i * din + 2 * hi;
    const float  ic   = inv[arow_i];

    const int colr = n0 + r;                     // < doutp always
    const float* wl0 = Wl + (long)(2 * hi) * doutp + colr;
    const float* wr0 = Wr + (long)(2 * hi) * doutp + colr;

    v8f acc0 = {}, acc1 = {}, acc2 = {}, acc3 = {};

#pragma unroll 2
    for (int k = 0; k < din; k += 4) {
        // A fragments (fp32, full precision)
        v2f ah = *(const v2f*)(hrow + k);
        v2f at = *(const v2f*)(arow + k);
        v2f aa;
        aa.x = at.x * ic;
        aa.y = at.y * ic;

        const float* wl = wl0 + (long)k * doutp;  // row (k + 2*hi) at col colr
        const float* wr = wr0 + (long)k * doutp;

        v2f bl0, bl1, bl2, bl3, br0, br1, br2, br3;
        bl0.x = wl[0];   bl0.y = wl[doutp];
        bl1.x = wl[16];  bl1.y = wl[doutp + 16];
        bl2.x = wl[32];  bl2.y = wl[doutp + 32];
        bl3.x = wl[48];  bl3.y = wl[doutp + 48];
        br0.x = wr[0];   br0.y = wr[doutp];
        br1.x = wr[16];  br1.y = wr[doutp + 16];
        br2.x = wr[32];  br2.y = wr[doutp + 32];
        br3.x = wr[48];  br3.y = wr[doutp + 48];

        acc0 = __builtin_amdgcn_wmma_f32_16x16x4_f32(false, aa, false, bl0, (short)0, acc0, false, false);
        acc0 = __builtin_amdgcn_wmma_f32_16x16x4_f32(false, ah, false, br0, (short)0, acc0, false, false);
        acc1 = __builtin_amdgcn_wmma_f32_16x16x4_f32(false, aa, false, bl1, (short)0, acc1, false, false);
        acc1 = __builtin_amdgcn_wmma_f32_16x16x4_f32(false, ah, false, br1, (short)0, acc1, false, false);
        acc2 = __builtin_amdgcn_wmma_f32_16x16x4_f32(false, aa, false, bl2, (short)0, acc2, false, false);
        acc2 = __builtin_amdgcn_wmma_f32_16x16x4_f32(false, ah, false, br2, (short)0, acc2, false, false);
        acc3 = __builtin_amdgcn_wmma_f32_16x16x4_f32(false, aa, false, bl3, (short)0, acc3, false, false);
        acc3 = __builtin_amdgcn_wmma_f32_16x16x4_f32(false, ah, false, br3, (short)0, acc3, false, false);
    }

    // Epilogue: bias + (optional) relu + masked store (mask only matters for dout=500).
    const bool v0 = (colr      < dout);
    const bool v1 = (colr + 16 < dout);
    const bool v2 = (colr + 32 < dout);
    const bool v3 = (colr + 48 < dout);

    const float b0 = bias[colr];
    const float b1 = bias[colr + 16];
    const float b2 = bias[colr + 32];
    const float b3 = bias[colr + 48];

    const int rbase = m0 + 8 * hi;           // rows rbase .. rbase+7
    float* obase = out + (long)rbase * dout;

#pragma unroll
    for (int vr = 0; vr < 8; ++vr) {
        if (rbase + vr >= Nn) break;
        float* orow = obase + (long)vr * dout;
        float e0 = acc0[vr] + b0;
        float e1 = acc1[vr] + b1;
        float e2 = acc2[vr] + b2;
        float e3 = acc3[vr] + b3;
        if (do_relu) {
            e0 = fmaxf(e0, 0.f); e1 = fmaxf(e1, 0.f);
            e2 = fmaxf(e2, 0.f); e3 = fmaxf(e3, 0.f);
        }
        if (v0) orow[colr]      = e0;
        if (v1) orow[colr + 16] = e1;
        if (v2) orow[colr + 32] = e2;
        if (v3) orow[colr + 48] = e3;
    }
}

// ---------------------------------------------------------------------------
// L2-normalize + relu for the 128-wide layer 4 output. One wave32 per row,
// 4 floats per lane, shuffle reduction across the wave.
// ---------------------------------------------------------------------------
__global__ void k_norm_relu_128(float* __restrict__ h, int Nn) {
    const int lane = threadIdx.x & 31;
    const int wv   = threadIdx.x >> 5;
    const int row  = blockIdx.x * (blockDim.x >> 5) + wv;
    if (row >= Nn) return;
    float4 v = *(float4*)(h + (long)row * 128 + lane * 4);
    float s = v.x * v.x + v.y * v.y + v.z * v.z + v.w * v.w;
#pragma unroll
    for (int off = 16; off > 0; off >>= 1) s += __shfl_xor(s, off, 32);
    const float scale = 1.0f / fmaxf(sqrtf(s), 1e-12f);
    v.x = fmaxf(v.x * scale, 0.f);
    v.y = fmaxf(v.y * scale, 0.f);
    v.z = fmaxf(v.z * scale, 0.f);
    v.w = fmaxf(v.w * scale, 0.f);
    *(float4*)(h + (long)row * 128 + lane * 4) = v;
}

// ---------------------------------------------------------------------------
// Host-side orchestration
// ---------------------------------------------------------------------------
extern "C" void kernel_launch(void* const* d_in, const int* in_sizes, int n_in,
                              void* d_out, int out_size, void* d_ws, size_t ws_size,
                              hipStream_t stream) {
    static const int dims[9] = {500, 256, 128, 128, 128, 128, 128, 256, 500};

    const float* x  = (const float*)d_in[0];
    const int*   ei = (const int*)d_in[1];
    const int E  = in_sizes[1] / 2;
    const int Nn = in_sizes[0] / 500;

    float* ws   = (float*)d_ws;
    float* inv  = ws;                              // Nn
    float* cnt  = ws + Nn;                         // Nn
    float* agg  = ws + 2 * (size_t)Nn;             // Nn*500 (max din)
    float* bufA = agg  + (size_t)Nn * 500;         // Nn*256
    float* bufB = bufA + (size_t)Nn * 256;         // Nn*256
    float* wlp  = bufB + (size_t)Nn * 256;         // 500*512 pad scratch
    float* wrp  = wlp  + (size_t)500 * 512;        // 500*512 pad scratch
    float* bp   = wrp  + (size_t)500 * 512;        // 512 pad scratch

    // ---- in-degree & reciprocal (shared by all layers) ----
    {
        long n4 = (long)Nn / 4;
        k_fill_zero<<<(int)((n4 + 255) / 256), 256, 0, stream>>>(cnt, n4);
        k_count_deg<<<(E + 255) / 256, 256, 0, stream>>>(ei, E, cnt);
        k_inv_deg<<<(Nn + 255) / 256, 256, 0, stream>>>(cnt, inv, Nn);
    }

    const float* hcur = x;
    for (int l = 0; l < 8; ++l) {
        const int din  = dims[l];
        const int dout = dims[l + 1];
        const int doutp = (dout + 63) & ~63;       // 256/128 stay, 500 -> 512
        const float* Wl = (const float*)d_in[2 + 3 * l];
        const float* b  = (const float*)d_in[3 + 3 * l];
        const float* Wr = (const float*)d_in[4 + 3 * l];

        float* hout;
        if (l == 7)            hout = (float*)d_out;
        else if ((l & 1) == 0) hout = bufA;
        else                   hout = bufB;

        // zero agg region (Nn * din floats, multiple of 4)
        long n4 = (long)Nn * din / 4;
        k_fill_zero<<<(int)((n4 + 255) / 256), 256, 0, stream>>>(agg, n4);

        // mean-aggregate numerator: scatter-add h[src] into agg[dst]
        int total = E * (din >> 2);
        k_scatter<<<(total + 255) / 256, 256, 0, stream>>>(ei, E, hcur, agg, din, total);

        // pad weights/bias only when dout isn't tile-aligned (layer 8: 500->512)
        const float* Wlu = Wl;
        const float* Wru = Wr;
        const float* bu  = b;
        if (doutp != dout) {
            int tw = din * doutp;
            k_pack_pad<<<(tw + 255) / 256, 256, 0, stream>>>(Wl, wlp, din, dout, doutp);
            k_pack_pad<<<(tw + 255) / 256, 256, 0, stream>>>(Wr, wrp, din, dout, doutp);
            k_pack_bias<<<(doutp + 255) / 256, 256, 0, stream>>>(b, bp, dout, doutp);
            Wlu = wlp; Wru = wrp; bu = bp;
        }

        // fused GEMM (+bias, +relu except normalize layer)
        const int mtiles = (Nn + 15) >> 4;
        const int ntiles = doutp >> 6;
        const long waves = (long)mtiles * ntiles;
        const int do_relu = (l == 3) ? 0 : 1;
        k_sage_gemm<<<(int)((waves + 7) / 8), 256, 0, stream>>>(
            hcur, agg, inv, Wlu, Wru, bu, hout, Nn, din, dout, doutp,
            mtiles, ntiles, do_relu);

        if (l == 3) {  // conv4: F.normalize then relu
            k_norm_relu_128<<<(Nn + 7) / 8, 256, 0, stream>>>(hout, Nn);
        }
        hcur = hout;
    }
}